// GCNBlock_33818572488736
// MI455X (gfx1250) — compile-verified
//
#include <hip/hip_runtime.h>
#include <math.h>
#include <stdint.h>

typedef __attribute__((ext_vector_type(2))) float v2f;
typedef __attribute__((ext_vector_type(8))) float v8f;

#define C 128      // channels (C_IN == C_OUT == 128)
#define TM 16      // output rows per block in the WMMA GEMM

// ---------------------------------------------------------------------------
// 1) Spectral norm: v = normalize(W^T u); sigma = ||W v|| (one power step).
//    W is 128x128 — one 128-thread block.
// ---------------------------------------------------------------------------
__global__ void spectral_sigma_kernel(const float* __restrict__ W,
                                      const float* __restrict__ u,
                                      float* __restrict__ inv_sigma) {
  __shared__ float red[C];
  __shared__ float vsh[C];
  const int t = threadIdx.x;

  // s_j = (W^T u)_j
  float s = 0.f;
  for (int i = 0; i < C; ++i) s += W[i * C + t] * u[i];
  red[t] = s * s;
  __syncthreads();
  for (int off = C / 2; off > 0; off >>= 1) {
    if (t < off) red[t] += red[t + off];
    __syncthreads();
  }
  const float nrm_v = sqrtf(red[0]);
  __syncthreads();
  vsh[t] = s / (nrm_v + 1e-12f);
  __syncthreads();

  // t_i = (W v)_i
  float tt = 0.f;
  for (int j = 0; j < C; ++j) tt += W[t * C + j] * vsh[j];
  red[t] = tt * tt;
  __syncthreads();
  for (int off = C / 2; off > 0; off >>= 1) {
    if (t < off) red[t] += red[t + off];
    __syncthreads();
  }
  if (t == 0) {
    const float nrm_t = sqrtf(red[0]);
    // sigma = u2 . (W v) = ||Wv||^2 / (||Wv|| + eps)
    const float sigma = (nrm_t * nrm_t) / (nrm_t + 1e-12f);
    inv_sigma[0] = 1.0f / sigma;
  }
}

// ---------------------------------------------------------------------------
// 2) h = (x @ W^T) * inv_sigma via V_WMMA_F32_16X16X4_F32.
//    Block = 256 threads (8 waves); block computes 16 rows x 128 cols.
//    x tile (16x128) staged in LDS via CDNA5 async global->LDS B128 copies
//    (ASYNCcnt path, no VGPR round-trip). Row stride 132 floats -> the 32
//    A-fragment lane reads hit 32 distinct banks. Each wave owns a 16-column
//    slice and issues 32 WMMAs over K=128.
//    Fragment layouts per CDNA5 ISA 7.12.2:
//      A 16x4 f32 : lanes 0-15 -> M=lane, VGPR{0,1}=K{0,1}; lanes 16-31 K{2,3}
//      B  4x16 f32: lanes 0-15 -> N=lane, VGPR{0,1}=K{0,1}; lanes 16-31 K{2,3}
//      D 16x16 f32: VGPR r -> (M=r, N=lane) / (M=r+8, N=lane-16)
// ---------------------------------------------------------------------------
__global__ void __launch_bounds__(256) gemm_wmma_kernel(
    const float* __restrict__ x, const float* __restrict__ W,
    const float* __restrict__ inv_sigma, float* __restrict__ h, int nrows) {
  __shared__ float xs[TM][C + 4];
  const int row0 = blockIdx.x * TM;
  if (row0 + TM > nrows) return;  // N is a multiple of 16 (50000 = 3125*16)

  // Async staging of the 16x128 x-tile: 512 x 16B chunks, 2 per thread.
  // global_load_async_to_lds_b128 writes LDS directly (ASYNCcnt-tracked).
  {
    const int t = threadIdx.x;
#pragma unroll
    for (int i = 0; i < 2; ++i) {
      const int idx4 = t + i * 256;        // 0..511
      const int r = idx4 >> 5;             // row 0..15
      const int c4 = idx4 & 31;            // float4 col 0..31
      const float* gp = x + (size_t)(row0 + r) * C + c4 * 4;
      // Low 32 bits of the shared-aperture flat address == LDS byte offset.
      const unsigned lds_off = (unsigned)(uintptr_t)&xs[r][c4 * 4];
      asm volatile("global_load_async_to_lds_b128 %0, %1, off"
                   :
                   : "v"(lds_off), "v"(gp)
                   : "memory");
    }
    asm volatile("s_wait_asynccnt 0x0" ::: "memory");
  }
  __syncthreads();

  const int lane = threadIdx.x & 31;
  const int wave = threadIdx.x >> 5;    // 0..7 -> 16-column slice
  const int col0 = wave * 16;
  const int half = lane >> 4;           // 0 | 1
  const int l16  = lane & 15;
  const int ko   = half * 2;            // K sub-offset 0 | 2

  // B[k][n] = (W^T)[k][n] = W[col0+n][k]
  const float* __restrict__ wrow = W + (size_t)(col0 + l16) * C;

  v8f acc = {};
#pragma unroll
  for (int k = 0; k < C; k += 4) {
    v2f a, b;
    a.x = xs[l16][k + ko];
    a.y = xs[l16][k + ko + 1];
    b.x = wrow[k + ko];
    b.y = wrow[k + ko + 1];
    acc = __builtin_amdgcn_wmma_f32_16x16x4_f32(
        /*neg_a=*/false, a, /*neg_b=*/false, b,
        /*c_mod=*/(short)0, acc, /*reuse_a=*/false, /*reuse_b=*/false);
  }

  const float inv_s = inv_sigma[0];
  float* __restrict__ hp = h + (size_t)(row0 + half * 8) * C + col0 + l16;
#pragma unroll
  for (int r = 0; r < 8; ++r) hp[(size_t)r * C] = acc[r] * inv_s;
}

// ---------------------------------------------------------------------------
// 3) Degree (dst side, +1 self loop), dinv = rsqrt(deg),
//    out init = b + dinv^2 * h   (self-loop term).
// ---------------------------------------------------------------------------
__global__ void deg_init_kernel(float* __restrict__ deg, int n) {
  const int i = blockIdx.x * blockDim.x + threadIdx.x;
  if (i < n) deg[i] = 1.0f;  // self loop
}

__global__ void deg_count_kernel(const int* __restrict__ dst,
                                 float* __restrict__ deg, int E) {
  const int e = blockIdx.x * blockDim.x + threadIdx.x;
  if (e < E) unsafeAtomicAdd(&deg[dst[e]], 1.0f);
}

__global__ void self_init_kernel(const float* __restrict__ deg,
                                 const float* __restrict__ h,
                                 const float* __restrict__ bvec,
                                 float* __restrict__ dinv,
                                 float* __restrict__ out, int n) {
  const int row = blockIdx.x * 2 + (threadIdx.x >> 7);
  const int c = threadIdx.x & (C - 1);
  if (row >= n) return;
  const float di = rsqrtf(deg[row]);
  if (c == 0) dinv[row] = di;
  out[(size_t)row * C + c] = bvec[c] + di * di * h[(size_t)row * C + c];
}

// ---------------------------------------------------------------------------
// 4) Edge scatter: one wave32 per edge; lane holds float4 (32*16B = 512B row).
//    Gathers hit L2 (h = 25.6MB), scatters are no-return f32 atomics in L2.
// ---------------------------------------------------------------------------
__global__ void edge_aggregate_kernel(const int* __restrict__ ei,
                                      const float* __restrict__ h,
                                      const float* __restrict__ dinv,
                                      float* __restrict__ out, int E) {
  const int lane = threadIdx.x & 31;
  const int e = (blockIdx.x * blockDim.x + threadIdx.x) >> 5;
  if (e >= E) return;
  const int src = ei[e];
  const int dst = ei[E + e];
  const float coef = dinv[src] * dinv[dst];
  const float4 v = ((const float4*)(h + (size_t)src * C))[lane];
  float* o = out + (size_t)dst * C + lane * 4;
  unsafeAtomicAdd(o + 0, v.x * coef);
  unsafeAtomicAdd(o + 1, v.y * coef);
  unsafeAtomicAdd(o + 2, v.z * coef);
  unsafeAtomicAdd(o + 3, v.w * coef);
}

// ---------------------------------------------------------------------------
// 5) InstanceNorm over features (per row) + LeakyReLU(0.2), in place.
//    One wave32 per row, float4 per lane, shuffle-tree reductions.
// ---------------------------------------------------------------------------
__global__ void norm_lrelu_kernel(float* __restrict__ out, int n) {
  const int lane = threadIdx.x & 31;
  const int row = (blockIdx.x * blockDim.x + threadIdx.x) >> 5;
  if (row >= n) return;
  float4* rowp = (float4*)(out + (size_t)row * C);
  float4 v = rowp[lane];
  float s = v.x + v.y + v.z + v.w;
  float ss = v.x * v.x + v.y * v.y + v.z * v.z + v.w * v.w;
#pragma unroll
  for (int o = 16; o > 0; o >>= 1) {
    s += __shfl_xor(s, o, 32);
    ss += __shfl_xor(ss, o, 32);
  }
  const float mu = s * (1.0f / (float)C);
  const float var = ss * (1.0f / (float)C) - mu * mu;
  const float rinv = rsqrtf(var + 1e-5f);
  float4 y;
  y.x = (v.x - mu) * rinv;
  y.y = (v.y - mu) * rinv;
  y.z = (v.z - mu) * rinv;
  y.w = (v.w - mu) * rinv;
  y.x = y.x >= 0.f ? y.x : 0.2f * y.x;
  y.y = y.y >= 0.f ? y.y : 0.2f * y.y;
  y.z = y.z >= 0.f ? y.z : 0.2f * y.z;
  y.w = y.w >= 0.f ? y.w : 0.2f * y.w;
  rowp[lane] = y;
}

// ---------------------------------------------------------------------------
extern "C" void kernel_launch(void* const* d_in, const int* in_sizes, int n_in,
                              void* d_out, int out_size, void* d_ws, size_t ws_size,
                              hipStream_t stream) {
  const float* x  = (const float*)d_in[0];   // [N, 128]
  const int*   ei = (const int*)d_in[1];     // [2, E]
  const float* W  = (const float*)d_in[2];   // [128, 128]
  const float* b  = (const float*)d_in[3];   // [128]
  const float* u  = (const float*)d_in[4];   // [128]
  float* out = (float*)d_out;

  const int N = in_sizes[0] / C;             // 50000
  const int E = in_sizes[1] / 2;             // 800000

  // Workspace layout: [inv_sigma pad 256B][h: N*C f32][deg: N f32][dinv: N f32]
  char* ws = (char*)d_ws;
  float* inv_sigma = (float*)ws;
  float* h    = (float*)(ws + 256);
  float* deg  = (float*)(ws + 256 + (size_t)N * C * sizeof(float));
  float* dinv = deg + N;

  spectral_sigma_kernel<<<1, C, 0, stream>>>(W, u, inv_sigma);

  gemm_wmma_kernel<<<(N + TM - 1) / TM, 256, 0, stream>>>(x, W, inv_sigma, h, N);

  deg_init_kernel<<<(N + 255) / 256, 256, 0, stream>>>(deg, N);
  deg_count_kernel<<<(E + 255) / 256, 256, 0, stream>>>(ei + E, deg, E);
  self_init_kernel<<<(N + 1) / 2, 256, 0, stream>>>(deg, h, b, dinv, out, N);

  // one wave (32 threads) per edge -> E*32 threads
  edge_aggregate_kernel<<<(E * 32 + 255) / 256, 256, 0, stream>>>(ei, h, dinv, out, E);

  // one wave per row -> N*32 threads
  norm_lrelu_kernel<<<(N * 32 + 255) / 256, 256, 0, stream>>>(out, N);
}